// SparseZeroPadding_1125281432062
// MI455X (gfx1250) — compile-verified
//
#include <hip/hip_runtime.h>

// SparseZeroPadding: out[row, :] = feat[in_idx[row], :] if in_idx[row] >= 0 else 0
//
// Pure bandwidth-bound row gather (zero FLOPs -> WMMA does not apply).
// Strategy for MI455X (23.3 TB/s HBM, 192 MB L2, wave32):
//   * 8 lanes x float4 per 128-byte row  -> one b128 load + one b128 store per lane,
//     each valid row's gather is exactly one 128B cacheline request.
//   * Non-temporal stores for the 512 MB streaming output and non-temporal loads
//     for the one-pass index array, so the ~256 MB gather table keeps the L2.
//   * Masked rows skip the feat load entirely (saves ~half the gather traffic)
//     and store zeros (harness poisons d_out, so zeros must be written).

constexpr int C_FEAT = 32;                 // floats per row
constexpr int VEC = 4;                     // float4 per lane
constexpr int SEGS = C_FEAT / VEC;         // 8 lanes per row
constexpr int BLOCK = 256;                 // 8 waves (wave32)

typedef __attribute__((ext_vector_type(4))) float v4f;

__global__ __launch_bounds__(BLOCK) void sparse_zero_pad_gather(
    const float* __restrict__ feat,
    const int*   __restrict__ in_idx,
    float*       __restrict__ out,
    int n_rows)
{
    const int tid = blockIdx.x * BLOCK + threadIdx.x;
    const int row = tid >> 3;        // tid / SEGS
    const int seg = tid & (SEGS - 1);
    if (row >= n_rows) return;

    // Index is consumed once -> non-temporal load (don't cache 16 MB of indices).
    // 8 lanes of a row read the same address; coalesces to one request.
    const int idx = __builtin_nontemporal_load(&in_idx[row]);

    v4f val = {0.0f, 0.0f, 0.0f, 0.0f};
    if (idx >= 0) {
        // Regular-temporal load: the 256 MB feat table nearly fits the 192 MB L2,
        // and random gather indices repeat, so L2 residency pays off.
        const v4f* __restrict__ src =
            reinterpret_cast<const v4f*>(feat + (size_t)idx * C_FEAT);
        val = src[seg];              // global_load_b128, 128B/row across 8 lanes
    }

    // Streaming 512 MB output, never re-read -> non-temporal 128-bit store
    // so it does not evict the gather working set from L2.
    v4f* __restrict__ dst = reinterpret_cast<v4f*>(out + (size_t)row * C_FEAT);
    __builtin_nontemporal_store(val, &dst[seg]);   // global_store_b128 ... nt
}

extern "C" void kernel_launch(void* const* d_in, const int* in_sizes, int n_in,
                              void* d_out, int out_size, void* d_ws, size_t ws_size,
                              hipStream_t stream)
{
    (void)in_sizes; (void)n_in; (void)d_ws; (void)ws_size;

    const float* feat   = reinterpret_cast<const float*>(d_in[0]);
    const int*   in_idx = reinterpret_cast<const int*>(d_in[1]);  // int per harness contract
    float*       out    = reinterpret_cast<float*>(d_out);

    const int n_rows = out_size / C_FEAT;              // 4,000,000
    const int total_threads = n_rows * SEGS;           // 32,000,000 (< 2^31)
    const int grid = (total_threads + BLOCK - 1) / BLOCK;

    sparse_zero_pad_gather<<<grid, BLOCK, 0, stream>>>(feat, in_idx, out, n_rows);
}